// GCNConv_39479339384913
// MI455X (gfx1250) — compile-verified
//
#include <hip/hip_runtime.h>

#define BATCH 4
#define FDIM  64

typedef __attribute__((ext_vector_type(2))) float v2f;
typedef __attribute__((ext_vector_type(8))) float v8f;

// ---------------------------------------------------------------------------
// Kernel 1: zero the aggregation workspace (harness poisons ws with 0xAA).
// ---------------------------------------------------------------------------
__global__ void gcn_zero_ws(float* __restrict__ p, long n) {
    long i = (long)blockIdx.x * blockDim.x + threadIdx.x;
    const long stride = (long)gridDim.x * blockDim.x;
    for (; i < n; i += stride) p[i] = 0.0f;
}

// ---------------------------------------------------------------------------
// Kernel 2: SpMM scatter.  agg[b, rows[e], f] += vals[e] * x[b, cols[e], f]
// One thread per (edge, feature); loops over the 4 batches.
// x reads are coalesced 64-wide per edge; x (51 MB) lives in the 192 MB L2,
// so the random gather is an L2 hit.  Atomic f32 adds go straight to L2.
// ---------------------------------------------------------------------------
__global__ void gcn_spmm_scatter(const float* __restrict__ x,
                                 const float* __restrict__ vals,
                                 const int*   __restrict__ rows,
                                 const int*   __restrict__ cols,
                                 float*       __restrict__ agg,
                                 long E, long Nn) {
    const long g = (long)blockIdx.x * blockDim.x + threadIdx.x;
    const long e = g >> 6;            // edge index
    if (e >= E) return;
    const int  f = (int)(g & 63);     // feature index
    const float v = vals[e];
    const long  r = rows[e];
    const long  c = cols[e];
#pragma unroll
    for (int b = 0; b < BATCH; ++b) {
        const float xv = x[((long)b * Nn + c) * FDIM + f];
        __hip_atomic_fetch_add(&agg[((long)b * Nn + r) * FDIM + f], v * xv,
                               __ATOMIC_RELAXED, __HIP_MEMORY_SCOPE_AGENT);
    }
}

// ---------------------------------------------------------------------------
// Kernel 3: dense GEMM  out = elu(agg @ W + bias)  via V_WMMA_F32_16X16X4_F32.
// Rows = B*N = 200000 (multiple of 16), K = 64, Ncols = 64.
// One wave -> 16x64 output tile: 4 accumulators (8 VGPRs each),
// 16 K-steps x 4 N-tiles = 64 WMMAs per wave.
//
// VGPR layouts (ISA 7.12.2, 32-bit data):
//   A 16x4 : lane 0-15: M=lane, {K=4k+0, 4k+1}; lane 16-31: M=lane-16, {4k+2, 4k+3}
//   B 4x16 : lane 0-15: N=lane, rows {4k+0, 4k+1}; lane 16-31: rows {4k+2, 4k+3}
//   C/D    : VGPR r: lane 0-15 -> M=r, lane 16-31 -> M=r+8; N = lane&15
// ---------------------------------------------------------------------------
__global__ void __launch_bounds__(256)
gcn_gemm_bias_elu(const float* __restrict__ agg,
                  const float* __restrict__ W,      // [64,64] row-major (k,n)
                  const float* __restrict__ bias,   // [64]
                  float*       __restrict__ out,
                  long nrows) {
    const int lane = threadIdx.x & 31;
    const int wave = threadIdx.x >> 5;
    const long rowBase = ((long)blockIdx.x * 8 + wave) * 16;
    if (rowBase >= nrows) return;          // wave-uniform: EXEC stays all-1s

    const int m  = lane & 15;              // M (for A) / N (for B,D) within tile
    const int kg = (lane >> 4) << 1;       // K sub-group: 0 or 2

    long arow = rowBase + m;
    if (arow >= nrows) arow = nrows - 1;   // clamp (never triggers: 200000 % 16 == 0)
    const float* __restrict__ aptr = agg + arow * FDIM;

    v8f acc[4] = {v8f{}, v8f{}, v8f{}, v8f{}};

#pragma unroll
    for (int k4 = 0; k4 < FDIM / 4; ++k4) {
        const int kk = k4 * 4 + kg;
        v2f a;
        a.x = aptr[kk];
        a.y = aptr[kk + 1];

        const float* __restrict__ w0 = W + (long)kk * FDIM + m;        // row kk
        const float* __restrict__ w1 = W + (long)(kk + 1) * FDIM + m;  // row kk+1
#pragma unroll
        for (int j = 0; j < 4; ++j) {
            v2f b;
            b.x = w0[j * 16];
            b.y = w1[j * 16];
            acc[j] = __builtin_amdgcn_wmma_f32_16x16x4_f32(
                /*neg_a=*/false, a, /*neg_b=*/false, b,
                /*c_mod=*/(short)0, acc[j],
                /*reuse_a=*/false, /*reuse_b=*/false);
        }
    }

    // Epilogue: bias + ELU, predicated stores (EXEC changes only after WMMAs).
    const int rowAdd = (lane >> 4) * 8;
#pragma unroll
    for (int j = 0; j < 4; ++j) {
        const int col = j * 16 + m;
        const float bb = bias[col];
#pragma unroll
        for (int r = 0; r < 8; ++r) {
            const long row = rowBase + rowAdd + r;
            if (row < nrows) {
                const float v = acc[j][r] + bb;
                out[row * FDIM + col] = (v > 0.0f) ? v : (__expf(v) - 1.0f);
            }
        }
    }
}

// ---------------------------------------------------------------------------
// Launcher.  Inputs (setup_inputs order): x, vals, weight, bias, rows, cols.
// d_ws holds the f32 aggregation buffer agg[B, N, 64] (51.2 MB).
// ---------------------------------------------------------------------------
extern "C" void kernel_launch(void* const* d_in, const int* in_sizes, int n_in,
                              void* d_out, int out_size, void* d_ws, size_t ws_size,
                              hipStream_t stream) {
    const float* x      = (const float*)d_in[0];
    const float* vals   = (const float*)d_in[1];
    const float* weight = (const float*)d_in[2];
    const float* bias   = (const float*)d_in[3];
    const int*   rows   = (const int*)d_in[4];
    const int*   cols   = (const int*)d_in[5];
    float*       out    = (float*)d_out;

    const long E  = in_sizes[1];                       // 850000 edges
    const long Nn = (long)in_sizes[0] / (BATCH * FDIM); // 50000 nodes
    const long aggElems = (long)BATCH * Nn * FDIM;      // 12.8M floats
    float* agg = (float*)d_ws;

    // 1) zero agg (grid-stride)
    {
        const int  threads = 256;
        long blocks = (aggElems + threads - 1) / threads;
        if (blocks > 32768) blocks = 32768;
        gcn_zero_ws<<<(unsigned)blocks, threads, 0, stream>>>(agg, aggElems);
    }
    // 2) scatter-add SpMM
    {
        const long work = E * FDIM;
        const int  threads = 256;
        const long blocks = (work + threads - 1) / threads;
        gcn_spmm_scatter<<<(unsigned)blocks, threads, 0, stream>>>(
            x, vals, rows, cols, agg, E, Nn);
    }
    // 3) WMMA GEMM + bias + ELU
    {
        const long nrows  = (long)BATCH * Nn;          // 200000
        const long blocks = (nrows + 127) / 128;       // 128 rows per 8-wave block
        gcn_gemm_bias_elu<<<(unsigned)blocks, 256, 0, stream>>>(
            agg, weight, bias, out, nrows);
    }
}